// TestModel_86122684219691
// MI455X (gfx1250) — compile-verified
//
#include <hip/hip_runtime.h>
#include <hip/hip_bf16.h>

typedef __attribute__((ext_vector_type(2))) float v2f;
typedef __attribute__((ext_vector_type(8))) float v8f;
typedef __attribute__((ext_vector_type(4))) int   v4i;

#define I0_CONST 5.6e-06f
#define CHUNK 512

// ---- CDNA5 async global<->LDS path (guarded; falls back to plain copies) ----
#if defined(__gfx1250__) && __has_builtin(__builtin_amdgcn_global_load_async_to_lds_b128) && \
    __has_builtin(__builtin_amdgcn_global_store_async_from_lds_b32)
#define USE_ASYNC_LDS 1
#else
#define USE_ASYNC_LDS 0
#endif

#if defined(__gfx1250__) && __has_builtin(__builtin_amdgcn_s_wait_asynccnt)
#define WAIT_ASYNC() __builtin_amdgcn_s_wait_asynccnt(0)
#else
#define WAIT_ASYNC() asm volatile("s_wait_asynccnt 0x0" ::: "memory")
#endif

// Pointer types matching the builtin prototypes (global = AS1, LDS = AS3).
typedef __attribute__((address_space(1))) v4i* gptr_v4i;
typedef __attribute__((address_space(3))) v4i* lptr_v4i;
typedef __attribute__((address_space(1))) int* gptr_i32;
typedef __attribute__((address_space(3))) int* lptr_i32;

// ---------------------------------------------------------------------------
// Kernel 1: c[t] = dot(x[t, 0:6], b) via V_WMMA_F32_16X16X4_F32.
// A-tile: 16 rows x 4 K (f32); K padded 6 -> 8, two chained WMMAs.
// B: b[k] broadcast across all N columns -> every lane's C VGPR g holds
// c[row M = g (+8 for hi lanes)] regardless of N-lane mapping.
// ---------------------------------------------------------------------------
__global__ void __launch_bounds__(256)
cdot_wmma_kernel(const float* __restrict__ x, const float* __restrict__ b,
                 float* __restrict__ c, int T) {
    const int lane  = threadIdx.x & 31;
    const int gwave = (int)((blockIdx.x * blockDim.x + threadIdx.x) >> 5);
    const int nwave = (int)((gridDim.x * blockDim.x) >> 5);
    const bool hi   = lane >= 16;
    const int  mrow = lane & 15;

    const float b0 = b[0], b1 = b[1], b2 = b[2];
    const float b3 = b[3], b4 = b[4], b5 = b[5];

    v2f Blo, Bhi;
    Blo.x = hi ? b2 : b0;
    Blo.y = hi ? b3 : b1;
    Bhi.x = hi ? 0.0f : b4;   // K=6,7 zero padding
    Bhi.y = hi ? 0.0f : b5;

    const int ntiles = T >> 4;
    for (int tile = gwave; tile < ntiles; tile += nwave) {
        const float* px = x + (size_t)(tile * 16 + mrow) * 6;
        v2f Alo = *(const v2f*)(px + (hi ? 2 : 0));   // 8B aligned: 6*row even
        v2f Ahi = *(const v2f*)(px + 4);
        if (hi) { Ahi.x = 0.0f; Ahi.y = 0.0f; }

        v8f acc = {};
        acc = __builtin_amdgcn_wmma_f32_16x16x4_f32(
            false, Alo, false, Blo, (short)0, acc, false, false);
        acc = __builtin_amdgcn_wmma_f32_16x16x4_f32(
            false, Ahi, false, Bhi, (short)0, acc, false, false);

        // Lane 0 holds rows 0..7, lane 16 rows 8..15: two B128 stores each.
        if ((lane & 15) == 0) {
            float4* pc4 = (float4*)(c + tile * 16 + (hi ? 8 : 0));
            float4 v0 = {acc[0], acc[1], acc[2], acc[3]};
            float4 v1 = {acc[4], acc[5], acc[6], acc[7]};
            pc4[0] = v0;
            pc4[1] = v1;
        }
    }
}

// ---------------------------------------------------------------------------
// Kernel 2: sequential nonlinear scan, 2-op critical cycle:
//   u = fma(c, s, 1-k);  i' = i*u
//   v = fma(-c, i, 1);   s' = s*v
//   out[t+1] = 1 - s'   (out[0] = i0)
// Wave 0 lane 0 = scanner; wave 1 = DMA wave (async global<->LDS ping-pong).
// ---------------------------------------------------------------------------
__global__ void __launch_bounds__(64)
scan_kernel(const float* __restrict__ c, float* __restrict__ out,
            const float* __restrict__ kp, int T) {
    __shared__ __align__(16) float cbuf[2][CHUNK];
    __shared__ __align__(16) float ybuf[2][CHUNK];

    const int tid  = threadIdx.x;
    const int wave = tid >> 5;
    const int lane = tid & 31;

    const int nsteps = T - 1;                       // steps consume c[0..T-2]
    const int nch    = (nsteps + CHUNK - 1) / CHUNK;

    const float kabs = fabsf(kp[0]);
    const float omk  = 1.0f - kabs;

    // Prologue: stage chunk 0; write out[0] = i0.
    if (wave == 1) {
        const float4* src = (const float4*)c;
        for (int j = lane; j < CHUNK / 4; j += 32)
            ((float4*)cbuf[0])[j] = src[j];
        if (lane == 0) out[0] = I0_CONST;
    }
    __syncthreads();

    float iv = I0_CONST;
    float sv = 1.0f - I0_CONST;

    for (int n = 0; n < nch; ++n) {
        const int p = n & 1;
        if (wave == 1) {
            if (n + 1 < nch) {
                // Full-chunk fill always in-bounds (c has T elements).
#if USE_ASYNC_LDS
                const char* gsrc = (const char*)(c + (size_t)(n + 1) * CHUNK) + lane * 16;
                char*       ldst = (char*)cbuf[(n + 1) & 1] + lane * 16;
                __builtin_amdgcn_global_load_async_to_lds_b128(
                    (gptr_v4i)(gsrc +    0), (lptr_v4i)(ldst +    0), 0, 0);
                __builtin_amdgcn_global_load_async_to_lds_b128(
                    (gptr_v4i)(gsrc +  512), (lptr_v4i)(ldst +  512), 0, 0);
                __builtin_amdgcn_global_load_async_to_lds_b128(
                    (gptr_v4i)(gsrc + 1024), (lptr_v4i)(ldst + 1024), 0, 0);
                __builtin_amdgcn_global_load_async_to_lds_b128(
                    (gptr_v4i)(gsrc + 1536), (lptr_v4i)(ldst + 1536), 0, 0);
#else
                const float4* src = (const float4*)(c + (size_t)(n + 1) * CHUNK);
                float4* dst = (float4*)cbuf[(n + 1) & 1];
                for (int j = lane; j < CHUNK / 4; j += 32) dst[j] = src[j];
#endif
            }
            if (n >= 1) {
                const int base = (n - 1) * CHUNK;
#if USE_ASYNC_LDS
                char*       gdst = (char*)(out + base + 1) + lane * 4;  // 4B aligned
                const char* lsrc = (const char*)ybuf[(n - 1) & 1] + lane * 4;
#define Y_ASTORE(J)                                                        \
    __builtin_amdgcn_global_store_async_from_lds_b32(                      \
        (gptr_i32)(gdst + (J) * 128), (lptr_i32)(lsrc + (J) * 128), 0, 0)
                Y_ASTORE(0);  Y_ASTORE(1);  Y_ASTORE(2);  Y_ASTORE(3);
                Y_ASTORE(4);  Y_ASTORE(5);  Y_ASTORE(6);  Y_ASTORE(7);
                Y_ASTORE(8);  Y_ASTORE(9);  Y_ASTORE(10); Y_ASTORE(11);
                Y_ASTORE(12); Y_ASTORE(13); Y_ASTORE(14); Y_ASTORE(15);
#undef Y_ASTORE
#else
                const float* yb = ybuf[(n - 1) & 1];
                for (int j = lane; j < CHUNK; j += 32)
                    out[base + 1 + j] = yb[j];
#endif
            }
#if USE_ASYNC_LDS
            // All async DMAs must land before LDS ping-pong buffers are reused.
            WAIT_ASYNC();
#endif
        } else if (lane == 0) {
            const int cnt = min(CHUNK, nsteps - n * CHUNK);
            #pragma unroll 8
            for (int j = 0; j < cnt; ++j) {
                const float ct = cbuf[p][j];
                const float u  = fmaf(ct, sv, omk);    // 1-k + c*s
                const float v  = fmaf(-ct, iv, 1.0f);  // 1 - c*i   (old i)
                iv = iv * u;
                sv = sv * v;
                ybuf[p][j] = 1.0f - sv;
            }
        }
        __syncthreads();
    }

    // Epilogue: drain the final y chunk (plain stores, count-guarded).
    if (wave == 1) {
        const int n    = nch - 1;
        const int base = n * CHUNK;
        const int cnt  = nsteps - base;
        const float* yb = ybuf[n & 1];
        for (int j = lane; j < cnt; j += 32)
            out[base + 1 + j] = yb[j];
    }
}

extern "C" void kernel_launch(void* const* d_in, const int* in_sizes, int n_in,
                              void* d_out, int out_size, void* d_ws, size_t ws_size,
                              hipStream_t stream) {
    const float* x = (const float*)d_in[0];   // [T, 6] f32
    const float* b = (const float*)d_in[1];   // [6, 1] f32
    const float* k = (const float*)d_in[2];   // [1, 1] f32
    float* out = (float*)d_out;               // [1, T] f32
    float* cws = (float*)d_ws;                // scratch: c[T]

    const int T = in_sizes[0] / 6;            // 262144

    // Phase 1: memory-bound matvec (WMMA f32); 64 blocks x 8 waves = 512 waves.
    cdot_wmma_kernel<<<64, 256, 0, stream>>>(x, b, cws, T);
    // Phase 2: latency-bound serial scan; 1 block, 2 waves (scanner + DMA wave).
    scan_kernel<<<1, 64, 0, stream>>>(cws, out, k, T);
}